// GraphModel_38714835206273
// MI455X (gfx1250) — compile-verified
//
#include <hip/hip_runtime.h>
#include <math.h>

#define N_NODES  100000
#define N_EDGES  1600000
#define D_INPUT  128
#define D_HID    16
#define N_HEADS  8
#define NCLS     40
#define N_GRAPHS 64
#define ALPHA    0.2f

typedef float v2f __attribute__((ext_vector_type(2)));
typedef float v8f __attribute__((ext_vector_type(8)));

// ---------------------------------------------------------------- utilities
__global__ void fill_kernel(float* __restrict__ p, float v, int n) {
    int i = blockIdx.x * blockDim.x + threadIdx.x;
    if (i < n) p[i] = v;
}

__device__ inline void atomicMaxF(float* addr, float val) {
    // ordered-int trick: single hardware atomic, monotone under races
    if (val >= 0.0f) atomicMax((int*)addr, __float_as_int(val));
    else             atomicMin((unsigned int*)addr, (unsigned int)__float_as_int(val));
}

__device__ inline float lrelu(float x) { return x >= 0.0f ? x : ALPHA * x; }

// ---------------------------------------------------------------- GEMM1: Wh1 = x @ W1  (100000x128 * 128x128)
// block = 256 threads = 8 waves; each wave owns one 16-column tile; 16 rows per block.
__global__ __launch_bounds__(256) void gemm1_wmma(const float* __restrict__ x,
                                                  const float* __restrict__ W,
                                                  float* __restrict__ out) {
    __shared__ float Wl[D_INPUT * 128];               // 64 KB of the 320 KB WGP LDS
    int tid = threadIdx.x;
    const float4* Wv  = (const float4*)W;
    float4*       Wlv = (float4*)Wl;
    for (int i = tid; i < D_INPUT * 128 / 4; i += 256) Wlv[i] = Wv[i];
    __syncthreads();

    const int wave = tid >> 5;            // 0..7 -> column tile
    const int lane = tid & 31;
    const int m    = lane & 15;           // row within A tile / col within B,C,D tile
    const int kh   = (lane >> 4) * 2;     // K sub-offset 0 or 2
    const int row0 = blockIdx.x * 16;
    const int col  = wave * 16 + m;

    const float* xrow = x + (size_t)(row0 + m) * D_INPUT;
    v8f acc = {};
    #pragma unroll 4
    for (int kk = 0; kk < D_INPUT; kk += 4) {
        v2f a, b;
        a.x = xrow[kk + kh];
        a.y = xrow[kk + kh + 1];
        b.x = Wl[(kk + kh)     * 128 + col];
        b.y = Wl[(kk + kh + 1) * 128 + col];
        acc = __builtin_amdgcn_wmma_f32_16x16x4_f32(false, a, false, b,
                                                    (short)0, acc, false, false);
    }
    const int rbase = row0 + ((lane >> 4) ? 8 : 0);
    #pragma unroll
    for (int v = 0; v < 8; ++v)
        out[(size_t)(rbase + v) * 128 + col] = acc[v];
}

// ---------------------------------------------------------------- GEMM2: Wh2 = h1 @ W2 (100000x128 * 128x40)
// block = 128 threads = 4 waves; waves 0..2 own the three 16-col tiles (cols padded to 48).
__global__ __launch_bounds__(128) void gemm2_wmma(const float* __restrict__ h,
                                                  const float* __restrict__ W,
                                                  float* __restrict__ out) {
    __shared__ float Wl[D_INPUT * 48];                // 24 KB, zero-padded cols 40..47
    int tid = threadIdx.x;
    for (int i = tid; i < D_INPUT * 48; i += 128) {
        int r = i / 48, c = i % 48;
        Wl[i] = (c < NCLS) ? W[r * NCLS + c] : 0.0f;
    }
    __syncthreads();

    const int wave = tid >> 5;
    const int lane = tid & 31;
    if (wave >= 3) return;
    const int m    = lane & 15;
    const int kh   = (lane >> 4) * 2;
    const int row0 = blockIdx.x * 16;
    const int col  = wave * 16 + m;

    const float* hrow = h + (size_t)(row0 + m) * D_INPUT;
    v8f acc = {};
    #pragma unroll 4
    for (int kk = 0; kk < D_INPUT; kk += 4) {
        v2f a, b;
        a.x = hrow[kk + kh];
        a.y = hrow[kk + kh + 1];
        b.x = Wl[(kk + kh)     * 48 + col];
        b.y = Wl[(kk + kh + 1) * 48 + col];
        acc = __builtin_amdgcn_wmma_f32_16x16x4_f32(false, a, false, b,
                                                    (short)0, acc, false, false);
    }
    if (col < NCLS) {
        const int rbase = row0 + ((lane >> 4) ? 8 : 0);
        #pragma unroll
        for (int v = 0; v < 8; ++v)
            out[(size_t)(rbase + v) * NCLS + col] = acc[v];
    }
}

// ---------------------------------------------------------------- attention scores, layer 1
__global__ void scores1_kernel(const float* __restrict__ Wh,
                               const float* __restrict__ a1,
                               float* __restrict__ es, float* __restrict__ ed) {
    int gid = blockIdx.x * blockDim.x + threadIdx.x;       // N*H
    if (gid >= N_NODES * N_HEADS) return;
    int n = gid >> 3, h = gid & 7;
    const float* w = Wh + (size_t)n * 128 + h * D_HID;
    const float* av = a1 + h * 2 * D_HID;
    float s = 0.f, d = 0.f;
    #pragma unroll
    for (int k = 0; k < D_HID; ++k) { s += w[k] * av[k]; d += w[k] * av[D_HID + k]; }
    es[gid] = s; ed[gid] = d;
}

// ---------------------------------------------------------------- segment max / exp-sum, layer 1 (E*H threads)
__global__ void segmax1_kernel(const int* __restrict__ ei,
                               const float* __restrict__ es, const float* __restrict__ ed,
                               float* __restrict__ mx) {
    int gid = blockIdx.x * blockDim.x + threadIdx.x;
    if (gid >= N_EDGES * N_HEADS) return;
    int e = gid >> 3, h = gid & 7;
    int src = ei[e], dst = ei[N_EDGES + e];
    float v = lrelu(es[src * N_HEADS + h] + ed[dst * N_HEADS + h]);
    atomicMaxF(&mx[dst * N_HEADS + h], v);
}

__global__ void fixmax_kernel(float* __restrict__ mx, int n) {
    int i = blockIdx.x * blockDim.x + threadIdx.x;
    if (i < n && mx[i] == -INFINITY) mx[i] = 0.0f;
}

__global__ void expsum1_kernel(const int* __restrict__ ei,
                               const float* __restrict__ es, const float* __restrict__ ed,
                               const float* __restrict__ mx, float* __restrict__ sm) {
    int gid = blockIdx.x * blockDim.x + threadIdx.x;
    if (gid >= N_EDGES * N_HEADS) return;
    int e = gid >> 3, h = gid & 7;
    int src = ei[e], dst = ei[N_EDGES + e];
    float v  = lrelu(es[src * N_HEADS + h] + ed[dst * N_HEADS + h]);
    float ex = __expf(v - mx[dst * N_HEADS + h]);
    atomicAdd(&sm[dst * N_HEADS + h], ex);
}

// ---------------------------------------------------------------- weighted scatter-aggregate, layer 1 (E*128 threads)
__global__ void agg1_kernel(const int* __restrict__ ei,
                            const float* __restrict__ es, const float* __restrict__ ed,
                            const float* __restrict__ mx, const float* __restrict__ sm,
                            const float* __restrict__ Wh, float* __restrict__ out) {
    int gid = blockIdx.x * blockDim.x + threadIdx.x;       // E * 128
    if (gid >= N_EDGES * 128) return;
    int e = gid >> 7, t = gid & 127, h = t >> 4;
    int src = ei[e], dst = ei[N_EDGES + e];
    int nh = dst * N_HEADS + h;
    float v   = lrelu(es[src * N_HEADS + h] + ed[nh]);
    float att = __expf(v - mx[nh]) / (sm[nh] + 1e-16f);
    atomicAdd(&out[(size_t)dst * 128 + t], att * Wh[(size_t)src * 128 + t]);
}

__global__ void elu_kernel(float* __restrict__ p, int n) {
    int i = blockIdx.x * blockDim.x + threadIdx.x;
    if (i >= n) return;
    float v = p[i];
    p[i] = v > 0.0f ? v : (__expf(v) - 1.0f);
}

// ---------------------------------------------------------------- layer 2 (single head, D=40)
__global__ void scores2_kernel(const float* __restrict__ Wh,
                               const float* __restrict__ a2,
                               float* __restrict__ es, float* __restrict__ ed) {
    int n = blockIdx.x * blockDim.x + threadIdx.x;
    if (n >= N_NODES) return;
    const float* w = Wh + (size_t)n * NCLS;
    float s = 0.f, d = 0.f;
    #pragma unroll
    for (int k = 0; k < NCLS; ++k) { s += w[k] * a2[k]; d += w[k] * a2[NCLS + k]; }
    es[n] = s; ed[n] = d;
}

__global__ void segmax2_kernel(const int* __restrict__ ei,
                               const float* __restrict__ es, const float* __restrict__ ed,
                               float* __restrict__ mx) {
    int e = blockIdx.x * blockDim.x + threadIdx.x;
    if (e >= N_EDGES) return;
    int src = ei[e], dst = ei[N_EDGES + e];
    atomicMaxF(&mx[dst], lrelu(es[src] + ed[dst]));
}

__global__ void expsum2_kernel(const int* __restrict__ ei,
                               const float* __restrict__ es, const float* __restrict__ ed,
                               const float* __restrict__ mx, float* __restrict__ sm) {
    int e = blockIdx.x * blockDim.x + threadIdx.x;
    if (e >= N_EDGES) return;
    int src = ei[e], dst = ei[N_EDGES + e];
    atomicAdd(&sm[dst], __expf(lrelu(es[src] + ed[dst]) - mx[dst]));
}

__global__ void agg2_kernel(const int* __restrict__ ei,
                            const float* __restrict__ es, const float* __restrict__ ed,
                            const float* __restrict__ mx, const float* __restrict__ sm,
                            const float* __restrict__ Wh, float* __restrict__ out) {
    int gid = blockIdx.x * blockDim.x + threadIdx.x;       // E * 40
    if (gid >= N_EDGES * NCLS) return;
    int e = gid / NCLS, c = gid % NCLS;
    int src = ei[e], dst = ei[N_EDGES + e];
    float v   = lrelu(es[src] + ed[dst]);
    float att = __expf(v - mx[dst]) / (sm[dst] + 1e-16f);
    atomicAdd(&out[(size_t)dst * NCLS + c], att * Wh[(size_t)src * NCLS + c]);
}

// ---------------------------------------------------------------- global mean pool
__global__ void pool_kernel(const int* __restrict__ batch,
                            const float* __restrict__ h2,
                            float* __restrict__ pool) {
    int gid = blockIdx.x * blockDim.x + threadIdx.x;       // N * 40
    if (gid >= N_NODES * NCLS) return;
    int n = gid / NCLS, c = gid % NCLS;
    atomicAdd(&pool[batch[n] * NCLS + c], h2[(size_t)n * NCLS + c]);
}

__global__ void count_kernel(const int* __restrict__ batch, float* __restrict__ cnt) {
    int n = blockIdx.x * blockDim.x + threadIdx.x;
    if (n < N_NODES) atomicAdd(&cnt[batch[n]], 1.0f);
}

__global__ void finalize_kernel(const float* __restrict__ pool, const float* __restrict__ cnt,
                                float* __restrict__ out) {
    int gid = blockIdx.x * blockDim.x + threadIdx.x;       // 64 * 40
    if (gid >= N_GRAPHS * NCLS) return;
    int g = gid / NCLS;
    out[gid] = pool[gid] / fmaxf(cnt[g], 1.0f);
}

// ---------------------------------------------------------------- launch
extern "C" void kernel_launch(void* const* d_in, const int* in_sizes, int n_in,
                              void* d_out, int out_size, void* d_ws, size_t ws_size,
                              hipStream_t stream) {
    const float* x   = (const float*)d_in[0];
    const float* W1  = (const float*)d_in[1];
    const float* a1  = (const float*)d_in[2];
    const float* W2  = (const float*)d_in[3];
    const float* a2  = (const float*)d_in[4];
    const int*   ei  = (const int*)d_in[5];    // [2, E]
    const int*   bat = (const int*)d_in[6];    // [N]
    float* out = (float*)d_out;

    // workspace layout (floats)
    float* ws   = (float*)d_ws;
    float* Wh1  = ws;                              // N*128
    float* h1   = Wh1 + (size_t)N_NODES * 128;     // N*128
    float* Wh2  = h1  + (size_t)N_NODES * 128;     // N*40
    float* h2   = Wh2 + (size_t)N_NODES * NCLS;    // N*40
    float* es1  = h2  + (size_t)N_NODES * NCLS;    // N*8
    float* ed1  = es1 + (size_t)N_NODES * N_HEADS;
    float* m1   = ed1 + (size_t)N_NODES * N_HEADS;
    float* s1   = m1  + (size_t)N_NODES * N_HEADS;
    float* es2  = s1  + (size_t)N_NODES * N_HEADS; // N
    float* ed2  = es2 + N_NODES;
    float* m2   = ed2 + N_NODES;
    float* s2   = m2  + N_NODES;
    float* pool = s2  + N_NODES;                   // 64*40
    float* cnt  = pool + N_GRAPHS * NCLS;          // 64

    const int T = 256;
    auto blks = [](long n, int t) { return (int)((n + t - 1) / t); };

    // init accumulators
    fill_kernel<<<blks((long)N_NODES * 128, T), T, 0, stream>>>(h1, 0.0f, N_NODES * 128);
    fill_kernel<<<blks((long)N_NODES * NCLS, T), T, 0, stream>>>(h2, 0.0f, N_NODES * NCLS);
    fill_kernel<<<blks((long)N_NODES * N_HEADS, T), T, 0, stream>>>(m1, -INFINITY, N_NODES * N_HEADS);
    fill_kernel<<<blks((long)N_NODES * N_HEADS, T), T, 0, stream>>>(s1, 0.0f, N_NODES * N_HEADS);
    fill_kernel<<<blks(N_NODES, T), T, 0, stream>>>(m2, -INFINITY, N_NODES);
    fill_kernel<<<blks(N_NODES, T), T, 0, stream>>>(s2, 0.0f, N_NODES);
    fill_kernel<<<blks(N_GRAPHS * NCLS + N_GRAPHS, T), T, 0, stream>>>(pool, 0.0f, N_GRAPHS * NCLS + N_GRAPHS);

    // ---- layer 1
    gemm1_wmma<<<N_NODES / 16, 256, 0, stream>>>(x, W1, Wh1);
    scores1_kernel<<<blks((long)N_NODES * N_HEADS, T), T, 0, stream>>>(Wh1, a1, es1, ed1);
    segmax1_kernel<<<blks((long)N_EDGES * N_HEADS, T), T, 0, stream>>>(ei, es1, ed1, m1);
    fixmax_kernel<<<blks((long)N_NODES * N_HEADS, T), T, 0, stream>>>(m1, N_NODES * N_HEADS);
    expsum1_kernel<<<blks((long)N_EDGES * N_HEADS, T), T, 0, stream>>>(ei, es1, ed1, m1, s1);
    agg1_kernel<<<blks((long)N_EDGES * 128, T), T, 0, stream>>>(ei, es1, ed1, m1, s1, Wh1, h1);
    elu_kernel<<<blks((long)N_NODES * 128, T), T, 0, stream>>>(h1, N_NODES * 128);

    // ---- layer 2
    gemm2_wmma<<<N_NODES / 16, 128, 0, stream>>>(h1, W2, Wh2);
    scores2_kernel<<<blks(N_NODES, T), T, 0, stream>>>(Wh2, a2, es2, ed2);
    segmax2_kernel<<<blks(N_EDGES, T), T, 0, stream>>>(ei, es2, ed2, m2);
    fixmax_kernel<<<blks(N_NODES, T), T, 0, stream>>>(m2, N_NODES);
    expsum2_kernel<<<blks(N_EDGES, T), T, 0, stream>>>(ei, es2, ed2, m2, s2);
    agg2_kernel<<<blks((long)N_EDGES * NCLS, T), T, 0, stream>>>(ei, es2, ed2, m2, s2, Wh2, h2);

    // ---- pooling
    pool_kernel<<<blks((long)N_NODES * NCLS, T), T, 0, stream>>>(bat, h2, pool);
    count_kernel<<<blks(N_NODES, T), T, 0, stream>>>(bat, cnt);
    finalize_kernel<<<blks(N_GRAPHS * NCLS, T), T, 0, stream>>>(pool, cnt, out);
}